// WaveletTransform_56453050139015
// MI455X (gfx1250) — compile-verified
//
#include <hip/hip_runtime.h>

// Wavelet pyramid on MI455X (gfx1250, wave32).
// Fused formulation: the 3-level blur/downsample cascade composes into direct
// per-region separable triangle filters on the ORIGINAL input (stride 2/4/8,
// taps [1,2,1]/4, [1..4..1]/16, [1..8..1]/64). Regions written are disjoint:
//   copy kernel : everything outside [:256,:256]
//   S=2 kernel  : [:256,:256] \ [:128,:128]
//   S=4 kernel  : [:128,:128] \ [:64,:64]
//   S=8 kernel  : [:64,:64]
// Each filter tile: horizontal pass into LDS, vertical banded-matrix pass via
// V_WMMA_F32_16X16X4_F32 accumulate chain. Memory-bound (~512MB -> ~22us).

typedef float v2f __attribute__((ext_vector_type(2)));
typedef float v8f __attribute__((ext_vector_type(8)));

#define IMG  512
#define IMG2 (IMG * IMG)

// Triangle band weight: g[d] = (min(d, T-1-d)+1)/((R+1)^2), zero outside [0,T).
__device__ __forceinline__ float bandw(int d, int T, float inv) {
  int t = min(d, T - 1 - d);               // negative iff d outside [0, T)
  return (t >= 0) ? (float)(t + 1) * inv : 0.0f;
}

template <int S, int R>
__global__ __launch_bounds__(32)
void wavelet_region_kernel(const float* __restrict__ x, float* __restrict__ out,
                           int nt, int sk) {
  constexpr int   T     = 2 * R + 1;          // taps
  constexpr int   KROWS = 15 * S + T;         // patch rows needed per 16-row tile
  constexpr int   KPAD  = (KROWS + 3) & ~3;   // pad K to multiple of 4 for WMMA
  constexpr float INV   = 1.0f / (float)((R + 1) * (R + 1));

  const int tile = blockIdx.x;
  const int ti = tile / nt, tj = tile % nt;
  if (ti < sk && tj < sk) return;             // wave-uniform skip of inner region
  const int bc   = blockIdx.y;
  const int lane = threadIdx.x;               // 0..31, one wave per block

  const int i0 = ti * 16, j0 = tj * 16;
  const int row0 = S * i0 - R;                // top patch row (may be negative)
  const int col0 = S * j0 - R;                // left patch col (may be negative)

  const float* src = x   + (size_t)bc * IMG2;
  float*       dst = out + (size_t)bc * IMG2;

  __shared__ float H[KPAD][16];               // horizontally filtered patch

  // ---- Stage 1: horizontal triangle filter (stride S) into LDS ----
  for (int idx = lane; idx < KPAD * 16; idx += 32) {
    const int r = idx >> 4, n = idx & 15;
    const int rr = row0 + r;
    float acc = 0.0f;
    if (rr >= 0 && rr < IMG) {                // pad rows beyond KROWS get zero A-weight
      const float* rowp = src + rr * IMG;
      const int cbase = col0 + S * n;
#pragma unroll
      for (int v = 0; v < T; ++v) {
        const int cc = cbase + v;             // upper bound provably <= 511
        const float w = (float)(((v < T - 1 - v) ? v : (T - 1 - v)) + 1) * INV;
        if (cc >= 0) acc += w * rowp[cc];
      }
    }
    H[r][n] = acc;
  }
  __syncthreads();

  // ---- Stage 2: Out(16x16) = A(16xK) * H(Kx16), A = vertical band matrix ----
  // A fragment layout (f32 16x16x4): lanes 0-15 -> M=lane, K = k0+{0,1};
  //                                  lanes 16-31 -> M=lane-16, K = k0+{2,3}.
  // B fragment: N = lane&15, VGPR0/1 hold rows K = k0+khalf+{0,1}.
  v8f c = {};
  const int m     = lane & 15;
  const int khalf = (lane >> 4) << 1;         // 0 or 2
  for (int kk0 = 0; kk0 < KPAD; kk0 += 4) {
    const int k0 = kk0 + khalf;
    v2f a, b;
    a.x = bandw(k0     - S * m, T, INV);      // A[m][k0]
    a.y = bandw(k0 + 1 - S * m, T, INV);      // A[m][k0+1]
    b.x = H[k0][m];                           // B[k0][n],   n = lane&15
    b.y = H[k0 + 1][m];                       // B[k0+1][n]
    c = __builtin_amdgcn_wmma_f32_16x16x4_f32(false, a, false, b, (short)0, c,
                                              false, false);
  }

  // ---- Store D: vgpr r holds Out[r + (lane<16 ? 0 : 8)][lane&15] ----
  const int mbase = (lane >> 4) << 3;
#pragma unroll
  for (int r = 0; r < 8; ++r) {
    dst[(size_t)(i0 + mbase + r) * IMG + (j0 + m)] = c[r];
  }
}

// Copy region outside [:256,:256]: rows 256..511 full width, plus rows 0..255
// cols 256..511. Fully coalesced float4 traffic. 49152 float4 per (b,c) image.
__global__ __launch_bounds__(256)
void ring_copy_kernel(const float4* __restrict__ x4, float4* __restrict__ out4) {
  const int bc = blockIdx.y;
  const int i4 = blockIdx.x * 256 + threadIdx.x;     // [0, 49152), exact
  const size_t base = (size_t)bc * (IMG2 / 4);
  int off;
  if (i4 < 32768) {                 // rows 256..511, 128 float4 per row
    const int y  = 256 + (i4 >> 7);
    const int xq = i4 & 127;
    off = y * 128 + xq;
  } else {                          // rows 0..255, float4 cols 64..127
    const int j  = i4 - 32768;
    const int y  = j >> 6;
    const int xq = 64 + (j & 63);
    off = y * 128 + xq;
  }
  out4[base + off] = x4[base + off];
}

extern "C" void kernel_launch(void* const* d_in, const int* in_sizes, int n_in,
                              void* d_out, int out_size, void* d_ws, size_t ws_size,
                              hipStream_t stream) {
  (void)in_sizes; (void)n_in; (void)out_size; (void)d_ws; (void)ws_size;
  const float* x  = (const float*)d_in[0];
  float*      out = (float*)d_out;
  const int BC = 8 * 32;

  // Untouched ring copy (disjoint from all filter writes).
  ring_copy_kernel<<<dim3(192, BC), 256, 0, stream>>>(
      (const float4*)x, (float4*)out);

  // Level-1 region [:256,:256] minus [:128,:128]  (16x16 tiles, skip inner 8x8)
  wavelet_region_kernel<2, 1><<<dim3(16 * 16, BC), 32, 0, stream>>>(x, out, 16, 8);
  // Level-2 region [:128,:128] minus [:64,:64]    (8x8 tiles, skip inner 4x4)
  wavelet_region_kernel<4, 3><<<dim3(8 * 8,  BC), 32, 0, stream>>>(x, out, 8, 4);
  // Level-3 region [:64,:64]                      (4x4 tiles)
  wavelet_region_kernel<8, 7><<<dim3(4 * 4,  BC), 32, 0, stream>>>(x, out, 4, 0);
}